// MappedConvolution_58463094833220
// MI455X (gfx1250) — compile-verified
//
#include <hip/hip_runtime.h>
#include <math.h>

typedef __attribute__((ext_vector_type(2))) float v2f;
typedef __attribute__((ext_vector_type(8))) float v8f;

#define BN   2
#define CIN  32
#define COUT 64
#define KS   9
#define HH   256
#define WW   512
#define OHH  256
#define OWW  512
#define HWSZ (HH * WW)            // 131072
#define KDIM (CIN * KS)           // 288
#define KPAIRS (KDIM / 2)         // 144

// ---------------------------------------------------------------------------
// Kernel 1: repack weight[o][c][k] (o*288 + c*9 + k) into B-fragment-friendly
// layout: wpair[kp][o] = { W[o][c0][kk], W[o][c0+1][kk] } with
// kp = kk*16 + c0/2 (so kdim = kk*32 + c runs fastest over channels).
// 144 * 64 float2 = 72 KB in d_ws. Done once per launch; L2-resident after.
// ---------------------------------------------------------------------------
__global__ void repack_weight_kernel(const float* __restrict__ weight,
                                     v2f* __restrict__ wpair) {
    int idx = blockIdx.x * blockDim.x + threadIdx.x;   // 0 .. 9215
    if (idx >= KPAIRS * COUT) return;
    int kp = idx >> 6;          // / 64
    int o  = idx & 63;
    int kk = kp >> 4;           // / 16
    int c0 = (kp & 15) * 2;
    v2f v;
    v.x = weight[o * KDIM + c0 * KS + kk];
    v.y = weight[o * KDIM + (c0 + 1) * KS + kk];
    wpair[idx] = v;
}

// ---------------------------------------------------------------------------
// Kernel 2: implicit GEMM with V_WMMA_F32_16X16X4_F32.
// One wave (32 lanes) owns a 16-pixel M-tile and all 64 output channels:
//   acc0..acc3 : four 16x16 f32 accumulators (N-tiles of 16 channels).
// K dim = 288 = 9 sample points x 32 channels; bilinear coefficients are
// computed once per sample point and reused across the 8 channel steps.
// A-fragment (16x4 f32): lane = (m = lane&15, k-half = lane>>4); each lane
// produces its two bilinear-sampled values directly in the WMMA layout.
// ---------------------------------------------------------------------------
__global__ __launch_bounds__(256)
void mapped_conv_wmma_kernel(const float* __restrict__ x,
                             const v2f* __restrict__ wpair,
                             const float* __restrict__ bias,
                             const float* __restrict__ smap,
                             float* __restrict__ out) {
    __shared__ float lds_t[8 * 1088];   // per-wave 16x64 tile, stride 68 (pad)

    const int lane    = threadIdx.x & 31;
    const int wid     = threadIdx.x >> 5;
    const int lane_lo = lane & 15;
    const int lane_hi = lane >> 4;

    const int tile = blockIdx.x * 8 + wid;       // 16-pixel M-tile
    const int p    = tile * 16 + lane_lo;        // this lane's A-row pixel
    const int b    = p / (OHH * OWW);
    const int hw   = p - b * (OHH * OWW);

    const float* xb = x + b * (CIN * HWSZ);      // batch base
    const float* sm = smap + hw * (KS * 2);      // per-pixel sample coords

    v8f acc0 = {}; v8f acc1 = {}; v8f acc2 = {}; v8f acc3 = {};

    for (int kk = 0; kk < KS; ++kk) {
        // ---- bilinear setup (per pixel, shared across all 32 channels) ----
        float sx = sm[kk * 2 + 0];
        float sy = sm[kk * 2 + 1];
        float bx = floorf(sx), by = floorf(sy);
        float fx = sx - bx,    fy = sy - by;
        int ix0 = (int)bx, iy0 = (int)by;
        int ix1 = ix0 + 1, iy1 = iy0 + 1;
        ix0 = min(max(ix0, 0), WW - 1);  ix1 = min(max(ix1, 0), WW - 1);
        iy0 = min(max(iy0, 0), HH - 1);  iy1 = min(max(iy1, 0), HH - 1);
        float w00 = (1.f - fx) * (1.f - fy);
        float w10 = fx * (1.f - fy);
        float w01 = (1.f - fx) * fy;
        float w11 = fx * fy;
        int o00 = iy0 * WW + ix0, o10 = iy0 * WW + ix1;
        int o01 = iy1 * WW + ix0, o11 = iy1 * WW + ix1;

        // B fragments for this kk: wpair[(kk*16 + cs*2 + lane_hi)*64 + n]
        const v2f* wrow = wpair + (kk * 16 + lane_hi) * 64 + lane_lo;

        #pragma unroll
        for (int cs = 0; cs < 8; ++cs) {       // channel steps of 4 (WMMA K)
            const float* p0 = xb + (cs * 4 + 2 * lane_hi) * HWSZ;
            const float* p1 = p0 + HWSZ;
            float a0 = fmaf(w00, p0[o00],
                       fmaf(w10, p0[o10],
                       fmaf(w01, p0[o01], w11 * p0[o11])));
            float a1 = fmaf(w00, p1[o00],
                       fmaf(w10, p1[o10],
                       fmaf(w01, p1[o01], w11 * p1[o11])));
            v2f a; a.x = a0; a.y = a1;

            const v2f* wp = wrow + cs * 128;   // += cs*2 pair-rows of 64
            v2f b0 = wp[0];
            v2f b1 = wp[16];
            v2f b2 = wp[32];
            v2f b3 = wp[48];

            acc0 = __builtin_amdgcn_wmma_f32_16x16x4_f32(
                       false, a, false, b0, (short)0, acc0, false, false);
            acc1 = __builtin_amdgcn_wmma_f32_16x16x4_f32(
                       false, a, false, b1, (short)0, acc1, false, false);
            acc2 = __builtin_amdgcn_wmma_f32_16x16x4_f32(
                       false, a, false, b2, (short)0, acc2, false, false);
            acc3 = __builtin_amdgcn_wmma_f32_16x16x4_f32(
                       false, a, false, b3, (short)0, acc3, false, false);
        }
    }

    // ---- bias add (column o = ntile*16 + lane_lo is constant per lane) ----
    float bv0 = bias[lane_lo];
    float bv1 = bias[16 + lane_lo];
    float bv2 = bias[32 + lane_lo];
    float bv3 = bias[48 + lane_lo];
    #pragma unroll
    for (int r = 0; r < 8; ++r) {
        acc0[r] += bv0; acc1[r] += bv1; acc2[r] += bv2; acc3[r] += bv3;
    }

    // ---- transpose through LDS so stores are contiguous along w ----
    float* lw = lds_t + wid * 1088;            // [m][o], stride 68
    #pragma unroll
    for (int r = 0; r < 8; ++r) {
        int m = r + 8 * lane_hi;               // D layout: VGPR r -> row m
        lw[m * 68 +  0 + lane_lo] = acc0[r];
        lw[m * 68 + 16 + lane_lo] = acc1[r];
        lw[m * 68 + 32 + lane_lo] = acc2[r];
        lw[m * 68 + 48 + lane_lo] = acc3[r];
    }
    __syncthreads();

    // tile base -> (b, h, w) : tiles never straddle a row (OW % 16 == 0)
    int pbase = tile * 16;
    int bb  = pbase / (OHH * OWW);
    int hwb = pbase - bb * (OHH * OWW);
    int hb  = hwb / OWW;
    int wb  = hwb - hb * OWW;
    float* obase = out + ((bb * COUT) * OHH + hb) * OWW + wb;

    #pragma unroll
    for (int it = 0; it < 8; ++it) {
        int q  = it * 32 + lane;               // 0..255 over (o, j-block)
        int o  = q >> 2;
        int j0 = (q & 3) * 4;
        float4 v;
        v.x = lw[(j0 + 0) * 68 + o];
        v.y = lw[(j0 + 1) * 68 + o];
        v.z = lw[(j0 + 2) * 68 + o];
        v.w = lw[(j0 + 3) * 68 + o];
        *(float4*)(obase + o * (OHH * OWW) + j0) = v;   // 16B-aligned b128
    }
}

// ---------------------------------------------------------------------------
extern "C" void kernel_launch(void* const* d_in, const int* in_sizes, int n_in,
                              void* d_out, int out_size, void* d_ws, size_t ws_size,
                              hipStream_t stream) {
    const float* x      = (const float*)d_in[0];
    const float* weight = (const float*)d_in[1];
    const float* bias   = (const float*)d_in[2];
    const float* smap   = (const float*)d_in[3];
    float* out   = (float*)d_out;
    v2f*   wpair = (v2f*)d_ws;                 // needs 72 KB of scratch

    repack_weight_kernel<<<(KPAIRS * COUT + 255) / 256, 256, 0, stream>>>(
        weight, wpair);

    const int tiles = (BN * OHH * OWW) / 16;   // 16384 M-tiles
    mapped_conv_wmma_kernel<<<tiles / 8, 256, 0, stream>>>(
        x, wpair, bias, smap, out);
}